// FisherYatesHead_28329604284862
// MI455X (gfx1250) — compile-verified
//
#include <hip/hip_runtime.h>
#include <hip/hip_bf16.h>

typedef float v2f __attribute__((ext_vector_type(2)));
typedef float v8f __attribute__((ext_vector_type(8)));

#define B_DIM 16
#define S_DIM 512
#define SM1   511
#define D_DIM 1024

// One wave computes one 16(p) x 16(batch) tile of vals[:, i, :] with f32 WMMA.
//   A (16x4, M=p, K=d)  <- W[i, p0+row, d0..d0+3]   (row-major, b64/lane)
//   B (4x16, K=d, N=b)  <- x[b, i, d0..d0+3]        (b64/lane)
//   C/D (16x16 f32)     -> out[b, i, p] + bias[i,p], masked to -inf if i+p>=512
__global__ __launch_bounds__(128) void tri_linear_wmma_f32(
    const float* __restrict__ x,     // [16,512,1024]
    const float* __restrict__ W,     // [511,512,1024]
    const float* __restrict__ bias,  // [511,512]
    float* __restrict__ out)         // [16,511,512]
{
    const int i    = blockIdx.x;           // position 0..510
    const int wave = threadIdx.x >> 5;     // 0..3
    const int lane = threadIdx.x & 31;
    const int p0   = blockIdx.y * 64 + wave * 16;

    const int nlo  = lane & 15;            // A: p-row select / B: batch select
    const int half = lane >> 4;            // K sub-pair: lanes 16-31 hold K+2,K+3

    const float ninf = -__builtin_inff();

    // Tile entirely in the masked region: just fill -inf (wave-uniform branch).
    if (i + p0 >= S_DIM) {
#pragma unroll
        for (int r = 0; r < 8; ++r) {
            const int p = p0 + r + 8 * half;
            out[((size_t)nlo * SM1 + i) * S_DIM + p] = ninf;
        }
        return;
    }

    // Per-lane row base pointers (each lane walks contiguous d with b64 loads).
    const float* aRow = W + ((size_t)i * S_DIM + (size_t)(p0 + nlo)) * D_DIM + 2 * half;
    const float* bRow = x + ((size_t)nlo * S_DIM + (size_t)i) * D_DIM + 2 * half;

    // Two accumulators to break the WMMA->WMMA dependence chain.
    v8f c0 = {};
    v8f c1 = {};

#pragma unroll 4
    for (int d0 = 0; d0 < D_DIM; d0 += 8) {
        v2f a0 = *(const v2f*)(aRow + d0);
        v2f b0 = *(const v2f*)(bRow + d0);
        v2f a1 = *(const v2f*)(aRow + d0 + 4);
        v2f b1 = *(const v2f*)(bRow + d0 + 4);
        c0 = __builtin_amdgcn_wmma_f32_16x16x4_f32(false, a0, false, b0,
                                                   (short)0, c0, false, false);
        c1 = __builtin_amdgcn_wmma_f32_16x16x4_f32(false, a1, false, b1,
                                                   (short)0, c1, false, false);
    }

    const v8f c = c0 + c1;

    // D layout: lane n (0-15) holds N=batch=n, VGPR r holds M=p-row r (+8 for
    // lanes 16-31). Add bias, apply triangular mask, store.
#pragma unroll
    for (int r = 0; r < 8; ++r) {
        const int p = p0 + r + 8 * half;
        const float v = c[r] + bias[(size_t)i * S_DIM + p];
        out[((size_t)nlo * SM1 + i) * S_DIM + p] = (i + p < S_DIM) ? v : ninf;
    }
}

extern "C" void kernel_launch(void* const* d_in, const int* in_sizes, int n_in,
                              void* d_out, int out_size, void* d_ws, size_t ws_size,
                              hipStream_t stream) {
    (void)in_sizes; (void)n_in; (void)d_ws; (void)ws_size; (void)out_size;
    const float* x    = (const float*)d_in[0];  // [16,512,1024]
    const float* W    = (const float*)d_in[1];  // [511,512,1024]
    const float* bias = (const float*)d_in[2];  // [511,512]
    float* out        = (float*)d_out;          // [16,511,512]

    dim3 grid(SM1, S_DIM / 64, 1);  // 511 positions x 8 p-blocks of 64
    dim3 block(128, 1, 1);          // 4 waves, one 16x16 tile each
    tri_linear_wmma_f32<<<grid, block, 0, stream>>>(x, W, bias, out);
}